// SSIMLoss_42082089566696
// MI455X (gfx1250) — compile-verified
//
#include <hip/hip_runtime.h>

typedef __attribute__((ext_vector_type(2))) float v2f;
typedef __attribute__((ext_vector_type(8))) float v8f;

#define IMG_H 512
#define IMG_W 512
#define BATCH 64

// 7-tap Gaussian, sigma = 1.5, normalized (matches reference _gauss_window 1-D factor)
__device__ __constant__ float GK[7] = {
    0.03663284536f, 0.11128075901f, 0.21674532341f, 0.27068215609f,
    0.21674532341f, 0.11128075901f, 0.03663284536f};

// ---------------------------------------------------------------------------
// Kernel 1: fused SSIM over 16x16 tiles.
//   - horizontal 7-tap pass (VALU) -> LDS (rows y0-3 .. y0+18, padded to 24)
//   - vertical 7-tap pass as banded matmul via V_WMMA_F32_16X16X4_F32
//   - elementwise SSIM + deterministic in-block reduction -> partial[block]
// blockDim = 256 (8 waves); each wave owns one 16x16 tile (block strip 16x128)
// grid = (4, 32, 64)  -> 8192 blocks
// ---------------------------------------------------------------------------
__global__ __launch_bounds__(256) void ssim_tile_kernel(
    const float* __restrict__ pred, const float* __restrict__ targ,
    float* __restrict__ partial) {
  // per-wave row-convolved strips: [wave][map][row 0..23][col 0..15]
  __shared__ float ldsR[8][5][24][16];
  __shared__ float wsum[8];

  const int lane = threadIdx.x & 31;
  const int wave = threadIdx.x >> 5;
  const int hi   = lane >> 4;    // 0 for lanes 0-15, 1 for lanes 16-31
  const int n    = lane & 15;    // column / N index inside tile

  const int b  = blockIdx.z;
  const int y0 = blockIdx.y * 16;
  const int x0 = blockIdx.x * 128 + wave * 16;

  const float* __restrict__ pbase = pred + (size_t)b * (IMG_H * IMG_W);
  const float* __restrict__ tbase = targ + (size_t)b * (IMG_H * IMG_W);

  // ---- Stage 1: horizontal Gaussian on the 5 source maps (p, t, p2, t2, pt)
  // rows r = 0..21 map to image rows y0-3+r; rows 22,23 are zero K-padding.
#pragma unroll
  for (int kk = 0; kk < 12; ++kk) {
    const int r = 2 * kk + hi;
    float sp = 0.f, st = 0.f, sp2 = 0.f, st2 = 0.f, spt = 0.f;
    const int y = y0 - 3 + r;
    if (r < 22 && y >= 0 && y < IMG_H) {
      const float* __restrict__ prow = pbase + (size_t)y * IMG_W;
      const float* __restrict__ trow = tbase + (size_t)y * IMG_W;
#pragma unroll
      for (int j = 0; j < 7; ++j) {
        const int x = x0 + n - 3 + j;
        float p = 0.f, t = 0.f;
        if (x >= 0 && x < IMG_W) { p = prow[x]; t = trow[x]; }
        const float w = GK[j];
        sp  += w * p;
        st  += w * t;
        sp2 += w * (p * p);
        st2 += w * (t * t);
        spt += w * (p * t);
      }
    }
    ldsR[wave][0][r][n] = sp;
    ldsR[wave][1][r][n] = st;
    ldsR[wave][2][r][n] = sp2;
    ldsR[wave][3][r][n] = st2;
    ldsR[wave][4][r][n] = spt;
  }
  __syncthreads();

  // ---- Stage 2: vertical Gaussian as Out = A(16x24) x R(24x16), f32 WMMA.
  // A[i][j] = GK[j-i] for 0 <= j-i <= 6 (banded Toeplitz), constant per wave.
  // f32 A fragment layout (16x4 per step): lanes 0-15 hold (M=lane, K=k0,k0+1),
  // lanes 16-31 hold (M=lane-16, K=k0+2,k0+3). B mirrors with N=lane&15.
  float ax[6], ay[6];
#pragma unroll
  for (int s = 0; s < 6; ++s) {
    const int k  = 4 * s + 2 * hi;
    const int d0 = k - n;
    const int d1 = k + 1 - n;
    ax[s] = (d0 >= 0 && d0 <= 6) ? GK[d0] : 0.f;
    ay[s] = (d1 >= 0 && d1 <= 6) ? GK[d1] : 0.f;
  }

  v8f acc[5] = {};  // mu_p, mu_t, E[p2], E[t2], E[pt]
#pragma unroll
  for (int s = 0; s < 6; ++s) {
    const int k = 4 * s + 2 * hi;
    v2f a;
    a.x = ax[s];
    a.y = ay[s];
#pragma unroll
    for (int mi = 0; mi < 5; ++mi) {
      v2f bf;
      bf.x = ldsR[wave][mi][k][n];
      bf.y = ldsR[wave][mi][k + 1][n];
      acc[mi] = __builtin_amdgcn_wmma_f32_16x16x4_f32(
          /*neg_a=*/false, a, /*neg_b=*/false, bf,
          /*c_mod=*/(short)0, acc[mi], /*reuse_a=*/false, /*reuse_b=*/false);
    }
  }

  // ---- Stage 3: elementwise SSIM on the D fragments + reduction
  constexpr float C1 = 1e-4f;  // 0.01^2
  constexpr float C2 = 9e-4f;  // 0.03^2
  float lsum = 0.f;
#pragma unroll
  for (int e = 0; e < 8; ++e) {
    const float mp  = acc[0][e];
    const float mt  = acc[1][e];
    const float ep2 = acc[2][e];
    const float et2 = acc[3][e];
    const float ept = acc[4][e];
    const float mp2 = mp * mp;
    const float mt2 = mt * mt;
    const float mpt = mp * mt;
    const float sp2 = ep2 - mp2;
    const float st2 = et2 - mt2;
    const float spt = ept - mpt;
    const float num = (2.f * mpt + C1) * (2.f * spt + C2);
    const float den = (mp2 + mt2 + C1) * (sp2 + st2 + C2);
    lsum += num / den;
  }
#pragma unroll
  for (int off = 16; off > 0; off >>= 1)
    lsum += __shfl_xor(lsum, off, 32);
  if (lane == 0) wsum[wave] = lsum;
  __syncthreads();
  if (threadIdx.x == 0) {
    float s = 0.f;
#pragma unroll
    for (int w = 0; w < 8; ++w) s += wsum[w];
    const int blk = (blockIdx.z * 32 + blockIdx.y) * 4 + blockIdx.x;
    partial[blk] = s;
  }
}

// ---------------------------------------------------------------------------
// Kernel 2: deterministic final reduction of 8192 per-block partials.
// ---------------------------------------------------------------------------
__global__ __launch_bounds__(256) void ssim_reduce_kernel(
    const float* __restrict__ partial, int nblk, float* __restrict__ out) {
  __shared__ float sh[256];
  float s = 0.f;
  for (int i = threadIdx.x; i < nblk; i += 256) s += partial[i];
  sh[threadIdx.x] = s;
  __syncthreads();
#pragma unroll
  for (int stride = 128; stride > 0; stride >>= 1) {
    if ((int)threadIdx.x < stride) sh[threadIdx.x] += sh[threadIdx.x + stride];
    __syncthreads();
  }
  if (threadIdx.x == 0)
    out[0] = 1.f - sh[0] * (1.f / (float)(BATCH * IMG_H * IMG_W));
}

extern "C" void kernel_launch(void* const* d_in, const int* in_sizes, int n_in,
                              void* d_out, int out_size, void* d_ws,
                              size_t ws_size, hipStream_t stream) {
  const float* pred = (const float*)d_in[0];
  const float* targ = (const float*)d_in[1];
  float* out = (float*)d_out;
  float* partial = (float*)d_ws;  // 8192 floats = 32 KB of scratch

  dim3 grid(4, 32, 64);  // 128-col strips x 16-row strips x batch
  ssim_tile_kernel<<<grid, 256, 0, stream>>>(pred, targ, partial);
  ssim_reduce_kernel<<<1, 256, 0, stream>>>(partial, 4 * 32 * 64, out);
}